// MultiHeadSelfTokenAttention_1125281432103
// MI455X (gfx1250) — compile-verified
//
#include <hip/hip_runtime.h>
#include <hip/hip_bf16.h>

#define HID      768
#define NHEADS   12
#define HP       16      // heads padded to WMMA M=16
#define TOK      128
#define NBS      256     // B*S
#define MASK_NEG (-10000.0f)
#define LN_EPS   1e-5f

typedef __attribute__((ext_vector_type(2))) float v2f;
typedef __attribute__((ext_vector_type(8))) float v8f;

__device__ __forceinline__ v8f wmma_f32(v2f a, v2f b, v8f c) {
    // D = A(16x4 f32) * B(4x16 f32) + C(16x16 f32)
    return __builtin_amdgcn_wmma_f32_16x16x4_f32(
        /*neg_a=*/false, a, /*neg_b=*/false, b,
        /*c_mod=*/(short)0, c, /*reuse_a=*/false, /*reuse_b=*/false);
}

// ---------------------------------------------------------------------------
// Kernel 1: per (b,s) pair -> q = X@Wq (WMMA), masked softmax over tokens,
//           pooled = alpha^T @ X (WMMA).  pooled layout: [head][bs][k]
// ---------------------------------------------------------------------------
__global__ __launch_bounds__(256) void k_pool(const float* __restrict__ hidden,
                                              const float* __restrict__ mask,
                                              const float* __restrict__ Wq,
                                              const float* __restrict__ bq,
                                              float* __restrict__ pooled) {
    __shared__ float wq_p[HID * HP];   // 48 KB: Wq zero-padded to 16 heads
    __shared__ float qa[TOK * HP];     //  8 KB: q logits, then alpha, [t][h]

    const int bs   = blockIdx.x;
    const int tid  = threadIdx.x;
    const int wave = tid >> 5;
    const int lane = tid & 31;
    const int m16  = lane & 15;
    const int kh   = lane >> 4;        // 0 or 1: which K-half of the WMMA tile

    const float* X = hidden + (size_t)bs * TOK * HID;

    // stage padded Wq in LDS
    for (int i = tid; i < HID * HP; i += 256) {
        int k = i >> 4, n = i & 15;
        wq_p[i] = (n < NHEADS) ? Wq[k * NHEADS + n] : 0.0f;
    }
    __syncthreads();

    // ---- Phase A: q = X @ Wq  (wave w owns tokens [16w, 16w+16)) ----
    {
        const int t0 = wave * 16;
        const float* xrow = X + (size_t)(t0 + m16) * HID;
        v8f acc = {};
#pragma unroll 4
        for (int k0 = 0; k0 < HID; k0 += 4) {
            const int ka = k0 + 2 * kh;
            v2f a, b;
            a.x = xrow[ka];
            a.y = xrow[ka + 1];
            b.x = wq_p[ka * HP + m16];
            b.y = wq_p[(ka + 1) * HP + m16];
            acc = wmma_f32(a, b, acc);
        }
        const float bqv = (m16 < NHEADS) ? bq[m16] : 0.0f;
#pragma unroll
        for (int i = 0; i < 8; ++i) {
            const int m = i + kh * 8;                  // token within tile
            qa[(t0 + m) * HP + m16] = acc[i] + bqv;    // head = m16
        }
    }
    __syncthreads();

    // ---- softmax over the token axis, one head per (wave, pass) ----
    for (int h = wave; h < HP; h += 8) {
        if (h < NHEADS) {
            float sc[4], e[4];
            float mx = -3.0e38f;
#pragma unroll
            for (int j = 0; j < 4; ++j) {
                const int t = lane + 32 * j;
                sc[j] = qa[t * HP + h] + mask[bs * TOK + t] * MASK_NEG;
                mx = fmaxf(mx, sc[j]);
            }
#pragma unroll
            for (int off = 16; off > 0; off >>= 1)
                mx = fmaxf(mx, __shfl_xor(mx, off, 32));
            float sum = 0.0f;
#pragma unroll
            for (int j = 0; j < 4; ++j) { e[j] = __expf(sc[j] - mx); sum += e[j]; }
#pragma unroll
            for (int off = 16; off > 0; off >>= 1)
                sum += __shfl_xor(sum, off, 32);
            const float inv = 1.0f / sum;
#pragma unroll
            for (int j = 0; j < 4; ++j)
                qa[(lane + 32 * j) * HP + h] = e[j] * inv;
        } else {
#pragma unroll
            for (int j = 0; j < 4; ++j)
                qa[(lane + 32 * j) * HP + h] = 0.0f;   // padded heads
        }
    }
    __syncthreads();

    // ---- Phase B: pooled = alpha^T @ X  (M=16 heads, K=128 tokens, N=768) ----
    for (int n0 = wave * 16; n0 < HID; n0 += 128) {
        v8f acc = {};
#pragma unroll 4
        for (int k0 = 0; k0 < TOK; k0 += 4) {
            const int ka = k0 + 2 * kh;
            v2f a, b;
            a.x = qa[ka * HP + m16];                   // alpha^T[head=m16][tok=ka]
            a.y = qa[(ka + 1) * HP + m16];
            b.x = X[(size_t)ka * HID + n0 + m16];      // X[tok][n]
            b.y = X[(size_t)(ka + 1) * HID + n0 + m16];
            acc = wmma_f32(a, b, acc);
        }
#pragma unroll
        for (int i = 0; i < 8; ++i) {
            const int m = i + kh * 8;                  // head
            if (m < NHEADS)
                pooled[((size_t)m * NBS + bs) * HID + n0 + m16] = acc[i];
        }
    }
}

// ---------------------------------------------------------------------------
// Kernel 2: per head h: res[:, h*64 : h*64+64] = pooled_h[256,768] @ Wv block
// grid (16 m-tiles, 12 heads), 4 waves, one 16x16 N-tile per wave
// ---------------------------------------------------------------------------
__global__ __launch_bounds__(128) void k_res(const float* __restrict__ pooled,
                                             const float* __restrict__ Wv,
                                             const float* __restrict__ bv,
                                             float* __restrict__ res) {
    const int mtile = blockIdx.x;     // 0..15 : (b,s) row tile
    const int h     = blockIdx.y;     // 0..11
    const int tid   = threadIdx.x;
    const int wave  = tid >> 5, lane = tid & 31;
    const int m16   = lane & 15, kh = lane >> 4;
    const int col   = h * 64 + wave * 16 + m16;

    const float* arow = pooled + ((size_t)h * NBS + mtile * 16 + m16) * HID;

    v8f acc = {};
#pragma unroll 4
    for (int k0 = 0; k0 < HID; k0 += 4) {
        const int ka = k0 + 2 * kh;
        v2f a, b;
        a.x = arow[ka];
        a.y = arow[ka + 1];
        b.x = Wv[(size_t)ka * HID + col];
        b.y = Wv[(size_t)(ka + 1) * HID + col];
        acc = wmma_f32(a, b, acc);
    }
    const float bvv = bv[col];
#pragma unroll
    for (int i = 0; i < 8; ++i) {
        const int m = i + kh * 8;
        res[(size_t)(mtile * 16 + m) * HID + col] = acc[i] + bvv;
    }
}

// ---------------------------------------------------------------------------
// Kernel 3: opre = res[256,768] @ Wo[768,768] + bo
// grid (16 m-tiles, 6 n-groups), 8 waves, one 16x16 N-tile per wave
// ---------------------------------------------------------------------------
__global__ __launch_bounds__(256) void k_out(const float* __restrict__ res,
                                             const float* __restrict__ Wo,
                                             const float* __restrict__ bo,
                                             float* __restrict__ opre) {
    const int mtile = blockIdx.x;     // 0..15
    const int tid   = threadIdx.x;
    const int wave  = tid >> 5, lane = tid & 31;
    const int m16   = lane & 15, kh = lane >> 4;
    const int col   = (blockIdx.y * 8 + wave) * 16 + m16;   // 0..767

    const float* arow = res + (size_t)(mtile * 16 + m16) * HID;

    v8f acc = {};
#pragma unroll 4
    for (int k0 = 0; k0 < HID; k0 += 4) {
        const int ka = k0 + 2 * kh;
        v2f a, b;
        a.x = arow[ka];
        a.y = arow[ka + 1];
        b.x = Wo[(size_t)ka * HID + col];
        b.y = Wo[(size_t)(ka + 1) * HID + col];
        acc = wmma_f32(a, b, acc);
    }
    const float bov = bo[col];
#pragma unroll
    for (int i = 0; i < 8; ++i) {
        const int m = i + kh * 8;
        opre[(size_t)(mtile * 16 + m) * HID + col] = acc[i] + bov;
    }
}

// ---------------------------------------------------------------------------
// Kernel 4: row LayerNorm over 768, one block per (b,s)
// ---------------------------------------------------------------------------
__global__ __launch_bounds__(256) void k_ln(const float* __restrict__ opre,
                                            const float* __restrict__ gamma,
                                            const float* __restrict__ beta,
                                            float* __restrict__ out) {
    __shared__ float s1[8], s2[8];
    const int bs = blockIdx.x, tid = threadIdx.x;
    const float* row = opre + (size_t)bs * HID;

    const float x0 = row[tid], x1 = row[tid + 256], x2 = row[tid + 512];
    float s  = x0 + x1 + x2;
    float sq = x0 * x0 + x1 * x1 + x2 * x2;
#pragma unroll
    for (int off = 16; off > 0; off >>= 1) {
        s  += __shfl_xor(s,  off, 32);
        sq += __shfl_xor(sq, off, 32);
    }
    if ((tid & 31) == 0) { s1[tid >> 5] = s; s2[tid >> 5] = sq; }
    __syncthreads();
    float ts = 0.0f, tsq = 0.0f;
#pragma unroll
    for (int i = 0; i < 8; ++i) { ts += s1[i]; tsq += s2[i]; }
    const float mu   = ts * (1.0f / HID);
    const float var  = tsq * (1.0f / HID) - mu * mu;
    const float rstd = rsqrtf(var + LN_EPS);

    float* orow = out + (size_t)bs * HID;
    orow[tid]       = (x0 - mu) * rstd * gamma[tid]       + beta[tid];
    orow[tid + 256] = (x1 - mu) * rstd * gamma[tid + 256] + beta[tid + 256];
    orow[tid + 512] = (x2 - mu) * rstd * gamma[tid + 512] + beta[tid + 512];
}

// ---------------------------------------------------------------------------
extern "C" void kernel_launch(void* const* d_in, const int* in_sizes, int n_in,
                              void* d_out, int out_size, void* d_ws, size_t ws_size,
                              hipStream_t stream) {
    const float* hidden = (const float*)d_in[0];
    const float* mask   = (const float*)d_in[1];
    const float* Wq     = (const float*)d_in[2];
    const float* bq     = (const float*)d_in[3];
    const float* Wv     = (const float*)d_in[4];
    const float* bv     = (const float*)d_in[5];
    const float* Wo     = (const float*)d_in[6];
    const float* bo     = (const float*)d_in[7];
    const float* gamma  = (const float*)d_in[8];
    const float* beta   = (const float*)d_in[9];
    float* out = (float*)d_out;

    // workspace: pooled [12][256][768] | res [256][768] | opre [256][768]
    float* pooled = (float*)d_ws;
    float* res    = pooled + (size_t)NHEADS * NBS * HID;
    float* opre   = res + (size_t)NBS * HID;

    k_pool<<<NBS, 256, 0, stream>>>(hidden, mask, Wq, bq, pooled);
    k_res <<<dim3(16, 12), 128, 0, stream>>>(pooled, Wv, bv, res);
    k_out <<<dim3(16, 6), 256, 0, stream>>>(res, Wo, bo, opre);
    k_ln  <<<NBS, 256, 0, stream>>>(opre, gamma, beta, out);
}